// MultiScaleGATDecoder_20229295964944
// MI455X (gfx1250) — compile-verified
//
#include <hip/hip_runtime.h>
#include <hip/hip_bf16.h>
#include <math.h>

#define NN 20000
#define NE 320000
#define NH 4

typedef float v2f __attribute__((ext_vector_type(2)));
typedef float v8f __attribute__((ext_vector_type(8)));

enum { MODE_STORE = 0, MODE_RELU = 1, MODE_SIGMUL = 2 };

// ---------------------------------------------------------------------------
// fp32 WMMA GEMM:  D[N x M] = epilogue( [A1|A2] @ W + bias )
//   A1: Nrows x K1 row-major packed; A2: Nrows x K2 (optional, K2==0 -> unused)
//   W : (K1+K2) x M row-major.  K1, K2, K1+K2 all multiples of 32 (guaranteed
//   by the network dims), so a 32-wide K chunk never straddles the A1/A2 seam.
// One wave -> 16 x (NJ*16) tile via V_WMMA_F32_16X16X4_F32; 4 waves/block.
// A/B chunk tiles staged in LDS with coalesced float4 global loads; B stored
// pair-interleaved so each lane's (W[k][c], W[k+1][c]) fragment is one b64 DS
// load.  No branches inside the WMMA loop; EXEC stays all-ones.
// ---------------------------------------------------------------------------
template <int NJ>
__global__ __launch_bounds__(128)
void gemm_f32_wmma(const float* __restrict__ A1, int K1,
                   const float* __restrict__ A2, int K2,
                   const float* __restrict__ W,
                   const float* __restrict__ bias,
                   float* __restrict__ D,
                   int Nrows, int M, int mode)
{
    constexpr int KC      = 32;          // K chunk
    constexpr int COLS    = NJ * 16;     // block column tile
    constexpr int LDSA_LD = KC + 4;      // padded A row (16B-aligned, bank-spread)

    __shared__ float lds_a[64 * LDSA_LD];
    __shared__ float lds_b[(KC / 2) * (2 * COLS)];

    const int tid    = threadIdx.x;
    const int lane   = tid & 31;
    const int wave   = tid >> 5;
    const int lhalf  = lane >> 4;        // 0: lanes 0-15, 1: lanes 16-31
    const int lmod   = lane & 15;
    const int rowblk = blockIdx.y * 64;
    const int row0   = rowblk + wave * 16;
    const int col0   = blockIdx.x * COLS;
    const int Ktot   = K1 + K2;

    v8f acc[NJ] = {};

    for (int k0 = 0; k0 < Ktot; k0 += KC) {
        const bool  seg1  = (k0 < K1);
        const float* Aseg = seg1 ? A1 : A2;
        const int   kbase = seg1 ? k0 : k0 - K1;
        const int   ldA   = seg1 ? K1 : K2;

        __syncthreads();                 // previous chunk fully consumed

        // ---- stage A: 64 rows x KC cols (coalesced float4 loads) ----
#pragma unroll
        for (int f = tid; f < 64 * KC / 4; f += 128) {
            int r  = f >> 3;             // KC/4 = 8 float4 per row
            int c4 = (f & 7) << 2;
            int ar = rowblk + r;
            if (ar >= Nrows) ar = Nrows - 1;   // clamp; stores are guarded
            const float4 v = *(const float4*)(Aseg + (size_t)ar * ldA + kbase + c4);
            *(float4*)&lds_a[r * LDSA_LD + c4] = v;
        }

        // ---- stage B: KC rows x COLS cols, pair-interleaved ----
#pragma unroll
        for (int f = tid; f < KC * COLS / 4; f += 128) {
            int r  = f / (COLS / 4);
            int c4 = (f % (COLS / 4)) << 2;
            const float4 v = *(const float4*)(W + (size_t)(k0 + r) * M + col0 + c4);
            float* bp = &lds_b[(r >> 1) * (2 * COLS) + (r & 1)];
            bp[2 * (c4 + 0)] = v.x;
            bp[2 * (c4 + 1)] = v.y;
            bp[2 * (c4 + 2)] = v.z;
            bp[2 * (c4 + 3)] = v.w;
        }

        __syncthreads();

        // ---- compute: 8 K-steps x NJ WMMAs, all DS b64 fragment loads ----
#pragma unroll
        for (int kc = 0; kc < KC; kc += 4) {
            v2f a = *(const v2f*)&lds_a[(wave * 16 + lmod) * LDSA_LD + kc + 2 * lhalf];
            const int t = (kc >> 1) + lhalf;     // W row-pair index
#pragma unroll
            for (int j = 0; j < NJ; ++j) {
                v2f b = *(const v2f*)&lds_b[t * (2 * COLS) + 2 * (j * 16 + lmod)];
                acc[j] = __builtin_amdgcn_wmma_f32_16x16x4_f32(
                    false, a, false, b, (short)0, acc[j], false, false);
            }
        }
    }

    // ---- epilogue: fragment element i of lane-half h -> row = row0 + i + 8h
#pragma unroll
    for (int j = 0; j < NJ; ++j) {
        const int c = col0 + j * 16 + lmod;
        const float bs = bias ? bias[c] : 0.0f;
#pragma unroll
        for (int i = 0; i < 8; ++i) {
            int r = row0 + i + 8 * lhalf;
            if (r >= Nrows) continue;
            float v = acc[j][i] + bs;
            float* dp = D + (size_t)r * M + c;
            if (mode == MODE_RELU)        v = fmaxf(v, 0.0f);
            else if (mode == MODE_SIGMUL) v = dp[0] * (1.0f / (1.0f + __expf(-v)));
            dp[0] = v;
        }
    }
}

// ---------------------------------------------------------------------------
// Fills
// ---------------------------------------------------------------------------
__global__ void fill_f32(float* p, float v, int n) {
    int i = blockIdx.x * blockDim.x + threadIdx.x;
    if (i < n) p[i] = v;
}
__global__ void fill_u32(unsigned* p, unsigned v, int n) {
    int i = blockIdx.x * blockDim.x + threadIdx.x;
    if (i < n) p[i] = v;
}

// ---------------------------------------------------------------------------
// Ordered-float encoding for atomicMax on unsigned
// ---------------------------------------------------------------------------
__device__ __forceinline__ unsigned f2ord(float f) {
    unsigned u = __float_as_uint(f);
    return (u & 0x80000000u) ? ~u : (u | 0x80000000u);
}
__device__ __forceinline__ float ord2f(unsigned e) {
    unsigned b = (e & 0x80000000u) ? (e & 0x7FFFFFFFu) : ~e;
    return __uint_as_float(b);
}

// ---------------------------------------------------------------------------
// Attention coefficients: a_src[n,h] = sum_c h[n,h,c]*att_src[h,c]
// ---------------------------------------------------------------------------
__global__ void attn_coeff(const float* __restrict__ h,
                           const float* __restrict__ att_s,
                           const float* __restrict__ att_d,
                           float* __restrict__ a_s, float* __restrict__ a_d,
                           int N, int C)
{
    int idx = blockIdx.x * blockDim.x + threadIdx.x;
    if (idx >= N * NH) return;
    int n = idx / NH, hh = idx - n * NH;
    const float* hp = h + (size_t)n * NH * C + (size_t)hh * C;
    float ss = 0.f, sd = 0.f;
    for (int c = 0; c < C; ++c) {
        float v = hp[c];
        ss += v * att_s[hh * C + c];
        sd += v * att_d[hh * C + c];
    }
    a_s[idx] = ss; a_d[idx] = sd;
}

// alpha = leaky_relu(a_src[src]+a_dst[dst], 0.2); segment-max into amax[dst]
__global__ void edge_alpha_max(const long long* __restrict__ src,
                               const long long* __restrict__ dst,
                               const float* __restrict__ a_s,
                               const float* __restrict__ a_d,
                               float* __restrict__ alpha,
                               unsigned* __restrict__ amax)
{
    int idx = blockIdx.x * blockDim.x + threadIdx.x;
    if (idx >= NE * NH) return;
    int e = idx / NH, hh = idx - e * NH;
    int s = (int)src[e], d = (int)dst[e];
    float a = a_s[s * NH + hh] + a_d[d * NH + hh];
    a = (a > 0.0f) ? a : 0.2f * a;
    alpha[idx] = a;
    atomicMax(&amax[d * NH + hh], f2ord(a));
}

// ea = exp(alpha - amax[dst]); denom[dst] += ea  (ea stored over alpha)
__global__ void edge_exp_sum(const long long* __restrict__ dst,
                             float* __restrict__ alpha,
                             const unsigned* __restrict__ amax,
                             float* __restrict__ denom)
{
    int idx = blockIdx.x * blockDim.x + threadIdx.x;
    if (idx >= NE * NH) return;
    int e = idx / NH, hh = idx - e * NH;
    int d = (int)dst[e];
    float ea = __expf(alpha[idx] - ord2f(amax[d * NH + hh]));
    alpha[idx] = ea;
    atomicAdd(&denom[d * NH + hh], ea);
}

// out[dst] += h[src] * (ea/(denom[dst]+1e-16)); float4 per thread
__global__ void edge_aggregate(const long long* __restrict__ src,
                               const long long* __restrict__ dst,
                               const float* __restrict__ h,
                               const float* __restrict__ ea,
                               const float* __restrict__ denom,
                               float* __restrict__ out,
                               int tgt, int C)
{
    long long idx = (long long)blockIdx.x * blockDim.x + threadIdx.x;
    int vecs = tgt >> 2;
    if (idx >= (long long)NE * vecs) return;
    int e  = (int)(idx / vecs);
    int v  = (int)(idx - (long long)e * vecs);
    int c4 = v << 2;
    int hh = c4 / C;
    int s = (int)src[e], d = (int)dst[e];
    float w = ea[e * NH + hh] / (denom[d * NH + hh] + 1e-16f);
    const float4 m = *((const float4*)(h + (size_t)s * tgt) + v);
    float* op = out + (size_t)d * tgt + c4;
    atomicAdd(op + 0, m.x * w);
    atomicAdd(op + 1, m.y * w);
    atomicAdd(op + 2, m.z * w);
    atomicAdd(op + 3, m.w * w);
}

// ---------------------------------------------------------------------------
// In-place ReLU(LayerNorm(x + addb)); one wave32 per row, D in {128,256}
// ---------------------------------------------------------------------------
__global__ __launch_bounds__(256)
void ln_relu(float* __restrict__ X, const float* __restrict__ addb,
             const float* __restrict__ g, const float* __restrict__ b,
             int N, int D)
{
    int row  = blockIdx.x * 8 + (threadIdx.x >> 5);
    int lane = threadIdx.x & 31;
    if (row >= N) return;
    float* xp = X + (size_t)row * D;
    const int cnt = D >> 5;              // 4 or 8 elements per lane
    float v[8];
    float s = 0.0f;
#pragma unroll 8
    for (int i = 0; i < cnt; ++i) {
        int c = lane + (i << 5);
        float t = xp[c];
        if (addb) t += addb[c];
        v[i] = t; s += t;
    }
#pragma unroll
    for (int o = 16; o > 0; o >>= 1) s += __shfl_xor(s, o, 32);
    float mu = s / (float)D;
    float q = 0.0f;
#pragma unroll 8
    for (int i = 0; i < cnt; ++i) { float d0 = v[i] - mu; q += d0 * d0; }
#pragma unroll
    for (int o = 16; o > 0; o >>= 1) q += __shfl_xor(q, o, 32);
    float inv = rsqrtf(q / (float)D + 1e-5f);
#pragma unroll 8
    for (int i = 0; i < cnt; ++i) {
        int c = lane + (i << 5);
        float t = (v[i] - mu) * inv * g[c] + b[c];
        xp[c] = fmaxf(t, 0.0f);
    }
}

// ---------------------------------------------------------------------------
// Host orchestration
// ---------------------------------------------------------------------------
static inline dim3 g1(long long n, int b) { return dim3((unsigned)((n + b - 1) / b)); }

static void launch_gemm(const float* A1, int K1, const float* A2, int K2,
                        const float* W, const float* bias, float* D,
                        int Nrows, int M, int mode, hipStream_t stream)
{
    dim3 blk(128);
    if (M % 64 == 0) {
        dim3 grid(M / 64, (Nrows + 63) / 64);
        gemm_f32_wmma<4><<<grid, blk, 0, stream>>>(A1, K1, A2, K2, W, bias, D, Nrows, M, mode);
    } else {   // M == 32
        dim3 grid(M / 32, (Nrows + 63) / 64);
        gemm_f32_wmma<2><<<grid, blk, 0, stream>>>(A1, K1, A2, K2, W, bias, D, Nrows, M, mode);
    }
}

extern "C" void kernel_launch(void* const* d_in, const int* in_sizes, int n_in,
                              void* d_out, int out_size, void* d_ws, size_t ws_size,
                              hipStream_t stream)
{
    const float* enc0 = (const float*)d_in[0];
    const float* enc1 = (const float*)d_in[1];
    const float* enc2 = (const float*)d_in[2];
    const long long* eidx = (const long long*)d_in[3];
    const long long* esrc = eidx;
    const long long* edst = eidx + NE;

    // Workspace layout (floats)
    float* ws      = (float*)d_ws;
    float* buf_h   = ws;                        // NN*256 : h feats, then aligned(*gate)
    float* buf_g   = buf_h  + (size_t)NN * 256; // NN*256 : GAT agg -> ln'd dec
    float* buf_x1  = buf_g  + (size_t)NN * 256; // NN*256 : layer0 output
    float* ebuf    = buf_x1 + (size_t)NN * 256; // NE*NH  : alpha -> ea
    float* hid     = ebuf   + (size_t)NE * NH;  // NN*64  : gate hidden
    float* a_srcb  = hid    + (size_t)NN * 64;  // NN*NH
    float* a_dstb  = a_srcb + (size_t)NN * NH;  // NN*NH
    unsigned* amax = (unsigned*)(a_dstb + (size_t)NN * NH); // NN*NH
    float* denom   = (float*)(amax + (size_t)NN * NH);      // NN*NH

    const int layer_in [2] = {512, 256};
    const int layer_tgt[2] = {256, 128};
    const int layer_enc[2] = {256, 128};
    const float* layer_x  [2] = {enc2, buf_x1};
    const float* layer_e  [2] = {enc1, enc0};
    float*       layer_dst[2] = {buf_x1, (float*)d_out};

    for (int L = 0; L < 2; ++L) {
        const int in_dim  = layer_in[L];
        const int tgt     = layer_tgt[L];
        const int enc_dim = layer_enc[L];
        const int C       = tgt / NH;
        void* const* P = d_in + 4 + 18 * L;
        const float* W_gat   = (const float*)P[0];
        const float* att_src = (const float*)P[1];
        const float* att_dst = (const float*)P[2];
        const float* b_gat   = (const float*)P[3];
        const float* ln_g    = (const float*)P[4];
        const float* ln_b    = (const float*)P[5];
        const float* fa_W    = (const float*)P[6];
        const float* fa_b    = (const float*)P[7];
        const float* fa_ln_g = (const float*)P[8];
        const float* fa_ln_b = (const float*)P[9];
        const float* ag_W1   = (const float*)P[10];
        const float* ag_b1   = (const float*)P[11];
        const float* ag_W2   = (const float*)P[12];
        const float* ag_b2   = (const float*)P[13];
        const float* fu_W    = (const float*)P[14];
        const float* fu_b    = (const float*)P[15];
        const float* fu_ln_g = (const float*)P[16];
        const float* fu_ln_b = (const float*)P[17];

        const float* x   = layer_x[L];
        const float* enc = layer_e[L];
        float*       dst = layer_dst[L];

        dim3 blk256(256);

        // 1) h = x @ W_gat                                      [NN, tgt]
        launch_gemm(x, in_dim, nullptr, 0, W_gat, nullptr, buf_h, NN, tgt, MODE_STORE, stream);

        // 2) attention coefficients
        attn_coeff<<<g1((long long)NN * NH, 256), blk256, 0, stream>>>(
            buf_h, att_src, att_dst, a_srcb, a_dstb, NN, C);

        // 3) init segment buffers (0x007FFFFF == ord(-inf))
        fill_u32<<<g1((long long)NN * NH, 256), blk256, 0, stream>>>(amax, 0x007FFFFFu, NN * NH);
        fill_f32<<<g1((long long)NN * NH, 256), blk256, 0, stream>>>(denom, 0.0f, NN * NH);
        fill_f32<<<g1((long long)NN * tgt, 256), blk256, 0, stream>>>(buf_g, 0.0f, NN * tgt);

        // 4) edge softmax (max, exp+sum), 5) weighted aggregation
        edge_alpha_max<<<g1((long long)NE * NH, 256), blk256, 0, stream>>>(
            esrc, edst, a_srcb, a_dstb, ebuf, amax);
        edge_exp_sum<<<g1((long long)NE * NH, 256), blk256, 0, stream>>>(
            edst, ebuf, amax, denom);
        edge_aggregate<<<g1((long long)NE * (tgt / 4), 256), blk256, 0, stream>>>(
            esrc, edst, buf_h, ebuf, denom, buf_g, tgt, C);

        // 6) dec = relu(LN(agg + b_gat)) in-place in buf_g
        ln_relu<<<g1(NN, 8), blk256, 0, stream>>>(buf_g, b_gat, ln_g, ln_b, NN, tgt);

        // 7) aligned = relu(LN(enc @ fa_W + fa_b)) -> buf_h
        launch_gemm(enc, enc_dim, nullptr, 0, fa_W, fa_b, buf_h, NN, tgt, MODE_STORE, stream);
        ln_relu<<<g1(NN, 8), blk256, 0, stream>>>(buf_h, nullptr, fa_ln_g, fa_ln_b, NN, tgt);

        // 8) hid = relu([enc|dec] @ ag_W1 + ag_b1)              [NN, tgt/4]
        launch_gemm(enc, enc_dim, buf_g, tgt, ag_W1, ag_b1, hid, NN, tgt / 4, MODE_RELU, stream);

        // 9) buf_h (aligned) *= sigmoid(hid @ ag_W2 + ag_b2)
        launch_gemm(hid, tgt / 4, nullptr, 0, ag_W2, ag_b2, buf_h, NN, tgt, MODE_SIGMUL, stream);

        // 10) out = relu(LN([aligned*gate | dec] @ fu_W + fu_b))
        launch_gemm(buf_h, tgt, buf_g, tgt, fu_W, fu_b, dst, NN, tgt, MODE_STORE, stream);
        ln_relu<<<g1(NN, 8), blk256, 0, stream>>>(dst, nullptr, fu_ln_g, fu_ln_b, NN, tgt);
    }
}